// RnnDecoder_34522947125820
// MI455X (gfx1250) — compile-verified
//
#include <hip/hip_runtime.h>
#include <hip/hip_bf16.h>

typedef __attribute__((ext_vector_type(2))) float v2f;
typedef __attribute__((ext_vector_type(8))) float v8f;

#define B_ 16
#define S_ 512
#define H_ 1024
#define E_ 256
#define V_ 50257

__device__ __forceinline__ v8f wmma4(v2f a, v2f b, v8f c) {
  // D = A(16x4) * B(4x16) + C, fp32, wave32
  return __builtin_amdgcn_wmma_f32_16x16x4_f32(
      /*neg_a=*/false, a, /*neg_b=*/false, b,
      /*c_mod=*/(short)0, c, /*reuse_a=*/false, /*reuse_b=*/false);
}

__device__ __forceinline__ float sigmoidf_(float x) {
  return 1.0f / (1.0f + expf(-x));
}

// -------- Kernel 1: dec_proj[b][n] = sum_k dec_hs[b][k] * w1_W[n][k] + w1_b[n]
// grid 64 (N tiles), block 32 (1 wave)
__global__ void k_dec_proj(const float* __restrict__ dec_hs,
                           const float* __restrict__ w1W,
                           const float* __restrict__ w1b,
                           float* __restrict__ dec_proj) {
  const int lane = threadIdx.x & 31;
  const int m = lane & 15;
  const int kh = (lane >> 4) << 1;          // 0 or 2
  const int n0 = blockIdx.x * 16;
  const float* arow = dec_hs + (size_t)m * H_;
  const float* brow = w1W + (size_t)(n0 + m) * H_;
  v8f acc = {};
#pragma unroll 4
  for (int k = 0; k < H_; k += 4) {
    v2f a = *(const v2f*)(arow + k + kh);
    v2f b = *(const v2f*)(brow + k + kh);
    acc = wmma4(a, b, acc);
  }
  const int nn = n0 + (lane & 15);
  const float bias = w1b[nn];
#pragma unroll
  for (int v = 0; v < 8; ++v) {
    const int mm = v + ((lane >> 4) << 3);
    dec_proj[(size_t)mm * H_ + nn] = acc[v] + bias;
  }
}

// -------- Kernel 2: fused  score[s][b] = va . tanh(dec_proj[b] + enc[s,b] @ w2^T + w2_b) + va_b
// grid 512 (one per s), block 512 (16 waves); each wave does 4 N-tiles of 16
__global__ void k_score(const float* __restrict__ enc,
                        const float* __restrict__ w2W,
                        const float* __restrict__ w2b,
                        const float* __restrict__ dec_proj,
                        const float* __restrict__ vaW,
                        const float* __restrict__ vab,
                        float* __restrict__ score /* (S,B) */) {
  __shared__ float red[16];
  const int s = blockIdx.x;
  const int tid = threadIdx.x;
  if (tid < 16) red[tid] = 0.0f;
  __syncthreads();

  const int lane = tid & 31;
  const int wave = tid >> 5;
  const int m = lane & 15;                  // batch row within tile AND B-col index
  const int kh = (lane >> 4) << 1;
  const float* arow = enc + ((size_t)s * B_ + m) * H_;   // (S,B,H) row
  const int mbase = (lane >> 4) << 3;

  for (int t = 0; t < 4; ++t) {
    const int n0 = (wave * 4 + t) * 16;
    const float* brow = w2W + (size_t)(n0 + m) * H_;
    v8f acc = {};
#pragma unroll 4
    for (int k = 0; k < H_; k += 4) {
      v2f a = *(const v2f*)(arow + k + kh);
      v2f b = *(const v2f*)(brow + k + kh);
      acc = wmma4(a, b, acc);
    }
    const int nn = n0 + m;
    const float wb = w2b[nn];
    const float va = vaW[nn];
#pragma unroll
    for (int v = 0; v < 8; ++v) {
      const int mm = v + mbase;             // batch index
      float u = tanhf(acc[v] + wb + dec_proj[(size_t)mm * H_ + nn]) * va;
      // reduce over the 16 n-columns held by this half-wave
      u += __shfl_xor(u, 1, 32);
      u += __shfl_xor(u, 2, 32);
      u += __shfl_xor(u, 4, 32);
      u += __shfl_xor(u, 8, 32);
      if ((lane & 15) == 0) atomicAdd(&red[mm], u);
    }
  }
  __syncthreads();
  if (tid < 16) score[(size_t)s * B_ + tid] = red[tid] + vab[0];
}

// -------- Kernel 3: softmax over S per batch. grid 16, block 256
__global__ void k_softmax(const float* __restrict__ score,
                          float* __restrict__ attn /* (B,S) */) {
  __shared__ float sm[256];
  const int b = blockIdx.x, tid = threadIdx.x;
  const float v0 = score[(size_t)tid * B_ + b];
  const float v1 = score[(size_t)(tid + 256) * B_ + b];
  sm[tid] = fmaxf(v0, v1);
  __syncthreads();
  for (int off = 128; off > 0; off >>= 1) {
    if (tid < off) sm[tid] = fmaxf(sm[tid], sm[tid + off]);
    __syncthreads();
  }
  const float mx = sm[0];
  __syncthreads();
  const float e0 = expf(v0 - mx), e1 = expf(v1 - mx);
  sm[tid] = e0 + e1;
  __syncthreads();
  for (int off = 128; off > 0; off >>= 1) {
    if (tid < off) sm[tid] += sm[tid + off];
    __syncthreads();
  }
  const float inv = 1.0f / sm[0];
  attn[(size_t)b * S_ + tid] = e0 * inv;
  attn[(size_t)b * S_ + tid + 256] = e1 * inv;
}

// -------- Kernel 4: context[b][h] = sum_s attn[b][s] * enc[s][b][h]. grid (16,8), block 128
__global__ void k_context(const float* __restrict__ enc,
                          const float* __restrict__ attn,
                          float* __restrict__ ctx) {
  __shared__ float a[S_];
  const int b = blockIdx.x;
  const int tid = threadIdx.x;
  for (int i = tid; i < S_; i += 128) a[i] = attn[(size_t)b * S_ + i];
  __syncthreads();
  const int h = blockIdx.y * 128 + tid;
  float acc = 0.0f;
#pragma unroll 4
  for (int s = 0; s < S_; ++s)
    acc += a[s] * enc[((size_t)s * B_ + b) * H_ + h];
  ctx[(size_t)b * H_ + h] = acc;
}

// -------- Kernel 5: gi[b][n] = [context|emb[x[b]]] @ W_ih^T + b_ih. grid 192, block 32
__global__ void k_gi(const float* __restrict__ ctx,
                     const float* __restrict__ emb,
                     const int* __restrict__ x,
                     const float* __restrict__ Wih,
                     const float* __restrict__ bih,
                     float* __restrict__ gi) {
  const int lane = threadIdx.x & 31;
  const int m = lane & 15;
  const int kh = (lane >> 4) << 1;
  const int n0 = blockIdx.x * 16;
  const float* actx = ctx + (size_t)m * H_;
  const float* aemb = emb + (size_t)x[m] * E_;
  const float* brow = Wih + (size_t)(n0 + m) * (H_ + E_);
  v8f acc = {};
#pragma unroll 4
  for (int k = 0; k < H_; k += 4) {
    v2f a = *(const v2f*)(actx + k + kh);
    v2f b = *(const v2f*)(brow + k + kh);
    acc = wmma4(a, b, acc);
  }
#pragma unroll 4
  for (int k = 0; k < E_; k += 4) {
    v2f a = *(const v2f*)(aemb + k + kh);
    v2f b = *(const v2f*)(brow + H_ + k + kh);
    acc = wmma4(a, b, acc);
  }
  const int nn = n0 + (lane & 15);
  const float bias = bih[nn];
#pragma unroll
  for (int v = 0; v < 8; ++v) {
    const int mm = v + ((lane >> 4) << 3);
    gi[(size_t)mm * (3 * H_) + nn] = acc[v] + bias;
  }
}

// -------- Kernel 6: GRU pointwise with h0 == 0  =>  gh == b_hh exactly.
// h = (1 - sigmoid(i_z + b_hh_z)) * tanh(i_n + sigmoid(i_r + b_hh_r) * b_hh_n)
// grid (16,4), block 256; writes new_dec_hs directly
__global__ void k_gru(const float* __restrict__ gi,
                      const float* __restrict__ bhh,
                      float* __restrict__ h /* (B,H) == new_dec_hs */) {
  const int b = blockIdx.x;
  const int j = blockIdx.y * 256 + threadIdx.x;
  const float ir = gi[(size_t)b * (3 * H_) + j] + bhh[j];
  const float iz = gi[(size_t)b * (3 * H_) + H_ + j] + bhh[H_ + j];
  const float in_ = gi[(size_t)b * (3 * H_) + 2 * H_ + j];
  const float r = sigmoidf_(ir);
  const float z = sigmoidf_(iz);
  const float n = tanhf(in_ + r * bhh[2 * H_ + j]);
  h[(size_t)b * H_ + j] = (1.0f - z) * n;
}

// -------- Kernel 7: fc_out[b][v] = h[b] . out_W[v] + out_b[v]
// ntiles = ceil(V/16) = 3142; 4 waves/block -> grid 786, block 128
__global__ void k_fc(const float* __restrict__ h,
                     const float* __restrict__ outW,
                     const float* __restrict__ outb,
                     float* __restrict__ fc) {
  const int lane = threadIdx.x & 31;
  const int wave = threadIdx.x >> 5;
  const int tile = blockIdx.x * 4 + wave;
  const int ntiles = (V_ + 15) / 16;
  if (tile >= ntiles) return;               // uniform per wave
  const int m = lane & 15;
  const int kh = (lane >> 4) << 1;
  const int n0 = tile * 16;
  const int nn = n0 + m;
  const int nr = (nn < V_) ? nn : (V_ - 1); // clamp row for tail tile
  const float* arow = h + (size_t)m * H_;
  const float* brow = outW + (size_t)nr * H_;
  v8f acc = {};
#pragma unroll 4
  for (int k = 0; k < H_; k += 4) {
    v2f a = *(const v2f*)(arow + k + kh);
    v2f b = *(const v2f*)(brow + k + kh);
    acc = wmma4(a, b, acc);
  }
  if (nn < V_) {
    const float bias = outb[nn];
#pragma unroll
    for (int v = 0; v < 8; ++v) {
      const int mm = v + ((lane >> 4) << 3);
      fc[(size_t)mm * V_ + nn] = acc[v] + bias;
    }
  }
}

extern "C" void kernel_launch(void* const* d_in, const int* in_sizes, int n_in,
                              void* d_out, int out_size, void* d_ws, size_t ws_size,
                              hipStream_t stream) {
  const int*   x      = (const int*)d_in[0];
  const float* dec_hs = (const float*)d_in[1];
  const float* enc    = (const float*)d_in[2];
  const float* emb    = (const float*)d_in[3];
  const float* w1W    = (const float*)d_in[4];
  const float* w1b    = (const float*)d_in[5];
  const float* w2W    = (const float*)d_in[6];
  const float* w2b    = (const float*)d_in[7];
  const float* vaW    = (const float*)d_in[8];
  const float* vab    = (const float*)d_in[9];
  const float* Wih    = (const float*)d_in[10];
  // d_in[11] = W_hh: provably unused because h0 == 0 in the reference
  const float* bih    = (const float*)d_in[12];
  const float* bhh    = (const float*)d_in[13];
  const float* outW   = (const float*)d_in[14];
  const float* outb   = (const float*)d_in[15];

  float* out  = (float*)d_out;
  float* fc   = out;                          // (B, V)
  float* ndh  = out + (size_t)B_ * V_;        // (1, B, H)
  float* attn = ndh + (size_t)B_ * H_;        // (B, S, 1)

  float* ws       = (float*)d_ws;
  float* dec_proj = ws;                                   // B*H
  float* score    = dec_proj + (size_t)B_ * H_;           // S*B
  float* context  = score + (size_t)S_ * B_;              // B*H
  float* gi       = context + (size_t)B_ * H_;            // B*3H

  k_dec_proj<<<64, 32, 0, stream>>>(dec_hs, w1W, w1b, dec_proj);
  k_score<<<S_, 512, 0, stream>>>(enc, w2W, w2b, dec_proj, vaW, vab, score);
  k_softmax<<<B_, 256, 0, stream>>>(score, attn);
  k_context<<<dim3(B_, 8), 128, 0, stream>>>(enc, attn, context);
  k_gi<<<192, 32, 0, stream>>>(context, emb, x, Wih, bih, gi);
  k_gru<<<dim3(B_, 4), 256, 0, stream>>>(gi, bhh, ndh);
  k_fc<<<786, 128, 0, stream>>>(ndh, outW, outb, fc);
}